// NeRF_PE_48619029791266
// MI455X (gfx1250) — compile-verified
//
#include <hip/hip_runtime.h>
#include <cstdint>

// NeRF positional encoding, MI455X (gfx1250).
// Store-bandwidth-bound: 192 MiB out @ 23.3 TB/s => ~8.6us floor.
// Per row (4 coords): compute the 64-float sin/cos block once (hw v_sin/v_cos,
// exact pow2 argument scaling), park it in a wave-private LDS slice, then
// DMA-replicate it 12x across the 768-wide output row with
// GLOBAL_STORE_ASYNC_FROM_LDS_B128 (ASYNCcnt). Double-buffered slices, no
// workgroup barriers: all hazards handled with wave-local s_wait_dscnt /
// s_wait_asynccnt.

#define HIDDEN          768
#define WAVES_PER_BLOCK 8
#define THREADS         (WAVES_PER_BLOCK * 32)
#define MAX_BLOCKS      2048

typedef int v4i __attribute__((ext_vector_type(4)));
typedef __attribute__((address_space(1))) v4i*  as1_v4i_p;
typedef __attribute__((address_space(3))) v4i*  as3_v4i_p;
typedef __attribute__((address_space(1))) void* as1_void_p;
typedef __attribute__((address_space(3))) void* as3_void_p;

__device__ __forceinline__ void async_store_b128_from_lds(float* gaddr, float* laddr) {
#if __has_builtin(__builtin_amdgcn_global_store_async_from_lds_b128)
    __builtin_amdgcn_global_store_async_from_lds_b128(
        (as1_v4i_p)(as1_void_p)gaddr,
        (as3_v4i_p)(as3_void_p)laddr,
        /*offset=*/0, /*cpol=*/0);
#else
    uint64_t ga = (uint64_t)gaddr;
    uint32_t la = (uint32_t)(uintptr_t)laddr;
    asm volatile("global_store_async_from_lds_b128 %0, %1, off"
                 :: "v"(ga), "v"(la) : "memory");
#endif
}

// Wait until <= N of this wave's async stores remain outstanding.
__device__ __forceinline__ void wait_asynccnt_le6() {
    asm volatile("" ::: "memory");
#if __has_builtin(__builtin_amdgcn_s_wait_asynccnt)
    __builtin_amdgcn_s_wait_asynccnt(6);
#else
    asm volatile("s_wait_asynccnt 6" ::: "memory");
#endif
    asm volatile("" ::: "memory");
}

// ds_store -> async-engine-reads-LDS hazard: DScnt and ASYNCcnt are separate
// queues with no mutual ordering, so drain DS before issuing async stores.
__device__ __forceinline__ void wait_dscnt0() {
    asm volatile("" ::: "memory");
#if __has_builtin(__builtin_amdgcn_s_wait_dscnt)
    __builtin_amdgcn_s_wait_dscnt(0);
#else
    asm volatile("s_wait_dscnt 0x0" ::: "memory");
#endif
    asm volatile("" ::: "memory");
}

__global__ __launch_bounds__(THREADS) void nerf_pe_kernel(
    const float* __restrict__ x,   // [nrows, 4]
    float* __restrict__ out,       // [nrows, 768]
    int nrows)
{
    // Two 64-float slices per wave (double buffer): 8 waves * 2 * 256B = 4 KB.
    __shared__ float lds[WAVES_PER_BLOCK * 2 * 64];

    const int lane   = threadIdx.x & 31;
    const int wave   = threadIdx.x >> 5;
    const int stride = gridDim.x * WAVES_PER_BLOCK;

    // Per-lane constants for block positions (lane) and (lane+32):
    // pos j: L=j>>3, k=j&7, c=k>>1, even k -> sin, odd k -> cos.
    // j and j+32 share k (and c); L differs by 4.
    const int   k   = lane & 7;
    const int   c   = k >> 1;
    const int   L1  = lane >> 3;                        // 0..3
    const float s1  = __int_as_float((126 + L1) << 23); // 2^(L1-1), exact
    const float s2  = s1 * 16.0f;                       // 2^(L1+3), exact
    const int   src = (lane * 4) & 63;                  // b128 source pos in block

    int buf = 0;
    for (int row = blockIdx.x * WAVES_PER_BLOCK + wave; row < nrows;
         row += stride, buf ^= 1) {
        float* slice = &lds[(wave * 2 + buf) * 64];

        // Slice reuse: the 6 async stores issued from this slice 2 iterations
        // ago must be done. Stores complete in order among stores, so
        // ASYNCcnt <= 6 (the previous iteration's stores) is sufficient.
        wait_asynccnt_le6();

        // ---- Phase 1: fill the 64-float block (2 unique values per lane).
        const float xv = x[row * 4 + c];
        // v_sin/v_cos take revolutions: sin(2^L*pi*x) = v_sin(x * 2^(L-1)).
        const float a1 = xv * s1;
        const float a2 = xv * s2;
        float v1, v2;
        if (k & 1) {
            v1 = __builtin_amdgcn_cosf(a1);
            v2 = __builtin_amdgcn_cosf(a2);
        } else {
            v1 = __builtin_amdgcn_sinf(a1);
            v2 = __builtin_amdgcn_sinf(a2);
        }
        slice[lane]      = v1;
        slice[lane + 32] = v2;

        wait_dscnt0();   // LDS data in place before the async engine reads it

        // ---- Phase 2: replicate block 12x = 6 chunks of 128 floats.
        // Chunk t: lane stores 16B from block pos (4*lane)%64 to
        // out[row*768 + t*128 + 4*lane].
        float* rowOut = out + (size_t)row * HIDDEN + lane * 4;
#pragma unroll
        for (int t = 0; t < 6; ++t) {
            async_store_b128_from_lds(rowOut + t * 128, slice + src);
        }
        // No drain here: S_ENDPGM performs an implicit wait-idle, and the
        // next iteration's wait_asynccnt_le6 covers slice reuse.
    }
}

extern "C" void kernel_launch(void* const* d_in, const int* in_sizes, int n_in,
                              void* d_out, int out_size, void* d_ws, size_t ws_size,
                              hipStream_t stream) {
    const float* x   = (const float*)d_in[0];
    float*       out = (float*)d_out;

    const int nrows = in_sizes[0] / 4;                 // 64*1024 = 65536
    int blocks = (nrows + WAVES_PER_BLOCK - 1) / WAVES_PER_BLOCK;
    if (blocks > MAX_BLOCKS) blocks = MAX_BLOCKS;      // grid-stride covers the rest

    nerf_pe_kernel<<<dim3(blocks), dim3(THREADS), 0, stream>>>(x, out, nrows);
}